// KPConvAggregateLayer_46136538694260
// MI455X (gfx1250) — compile-verified
//
#include <hip/hip_runtime.h>

// Problem constants (from the reference)
#define B_      8
#define N_      65536
#define CIN_    128
#define COUT_   128
#define KP_     15
#define KP_EXT  0.48f

// Tiling
#define NSPLIT_ 32               // slabs along n per batch
#define SLAB_   (N_ / NSPLIT_)   // 2048 points per workgroup
#define NT_     64               // points per staged tile
#define TILES_  (SLAB_ / NT_)    // 32 tiles per slab
#define PAD_    4
#define SROW_   (NT_ + PAD_)     // 68 words; 68 % 64 == 4 -> conflict-free b64 pattern

typedef float v2f __attribute__((ext_vector_type(2)));
typedef float v8f __attribute__((ext_vector_type(8)));
typedef __attribute__((address_space(3))) float lds_float;

__device__ __forceinline__ void async_wait0() {
#if __has_builtin(__builtin_amdgcn_s_wait_asynccnt)
    __builtin_amdgcn_s_wait_asynccnt(0);
#else
    asm volatile("s_wait_asynccnt 0x0" ::: "memory");
#endif
}

// 16B global -> LDS DMA, tracked by ASYNCcnt (CDNA5 GLOBAL_LOAD_ASYNC_TO_LDS_B128).
__device__ __forceinline__ void async_copy16(float* lds_dst, const float* gsrc) {
    unsigned ldsoff = (unsigned)(size_t)(lds_float*)lds_dst;  // LDS byte address
    asm volatile("global_load_async_to_lds_b128 %0, %1, off"
                 :: "v"(ldsoff), "v"(gsrc) : "memory");
}

// ---------------------------------------------------------------------------
// Pass 1: per-(batch, slab) partial wf[16][128] via V_WMMA_F32_16X16X4_F32.
//   M = kernel-point index (15 + 1 zero pad), N = 16-channel tile (1/wave),
//   inner K = points n in chunks of 4. Double-buffered async LDS staging.
// ---------------------------------------------------------------------------
__global__ __launch_bounds__(256) void kpconv_partial_kernel(
    const float* __restrict__ p,        // [B, N, 3]
    const float* __restrict__ x,        // [B, CIN, N]
    const float* __restrict__ kp,       // [K, 3]
    float* __restrict__ wfp)            // [B, NSPLIT, 16, 128] partials
{
    __shared__ float sW[2][16 * SROW_];     // W^T tile: [k][n_local]
    __shared__ float sX[2][CIN_ * SROW_];   // x tile:   [c][n_local]

    const int b    = blockIdx.x / NSPLIT_;
    const int slab = blockIdx.x % NSPLIT_;
    const int t    = threadIdx.x;
    const int lane = t & 31;
    const int wave = t >> 5;
    const int cw   = wave * 16;             // channel-tile base for this wave

    // Per-lane WMMA operand addressing (16x4 f32 A layout, mirrored for B):
    // lanes 0-15 hold inner k = {0,1}, lanes 16-31 hold inner k = {2,3}.
    const int mrow = lane & 15;
    const int koff = (lane >> 4) * 2;

    // W-staging assignment: 256 threads cover 64 n x 4 k-groups
    const int wn = t & 63;                  // n within tile
    const int kq = t >> 6;                  // k group: handles k = 4*kq .. 4*kq+3

    const float inv_ext = 1.0f / KP_EXT;
    const float* xb = x + (size_t)b * CIN_ * N_;
    const float* pb = p + (size_t)b * N_ * 3;
    const int nbase = slab * SLAB_;

    // ---- stage helpers -----------------------------------------------------
    auto stage_x = [&](int n0, int buf) {
        #pragma unroll
        for (int it = 0; it < 8; ++it) {
            const int e  = t + 256 * it;    // 2048 float4 elements total
            const int r  = e >> 4;          // channel row
            const int c4 = e & 15;          // float4 column
            async_copy16(&sX[buf][r * SROW_ + c4 * 4],
                         xb + (size_t)r * N_ + n0 + c4 * 4);
        }
    };
    auto stage_w = [&](int n0, int buf) {
        const float* pp = pb + (size_t)(n0 + wn) * 3;
        const float px = pp[0], py = pp[1], pz = pp[2];
        #pragma unroll
        for (int kk = 0; kk < 4; ++kk) {
            const int k = kq * 4 + kk;
            float w = 0.0f;
            if (k < KP_) {
                const float dx = px - kp[k * 3 + 0];
                const float dy = py - kp[k * 3 + 1];
                const float dz = pz - kp[k * 3 + 2];
                const float sq = dx * dx + dy * dy + dz * dz;
                w = fmaxf(1.0f - sqrtf(sq) * inv_ext, 0.0f);
            }
            sW[buf][k * SROW_ + wn] = w;    // row 15 = zero pad
        }
    };

    v8f acc = {0.f, 0.f, 0.f, 0.f, 0.f, 0.f, 0.f, 0.f};

    // ---- prologue: fill buffer 0 ------------------------------------------
    stage_x(nbase, 0);
    stage_w(nbase, 0);
    async_wait0();
    __syncthreads();

    for (int tile = 0; tile < TILES_; ++tile) {
        const int cur = tile & 1;

        // Prefetch next tile into the other buffer (DMA overlaps WMMA below)
        if (tile + 1 < TILES_) {
            const int n1 = nbase + (tile + 1) * NT_;
            stage_x(n1, cur ^ 1);
            stage_w(n1, cur ^ 1);
        }

        // ---- 16 WMMA steps over the current 64-point tile ----
        const float* aBase = &sW[cur][mrow * SROW_ + koff];
        const float* bBase = &sX[cur][(cw + mrow) * SROW_ + koff];
        #pragma unroll
        for (int s = 0; s < NT_ / 4; ++s) {
            const v2f a  = *(const v2f*)(aBase + s * 4);   // ds_load_b64
            const v2f bv = *(const v2f*)(bBase + s * 4);   // ds_load_b64
            acc = __builtin_amdgcn_wmma_f32_16x16x4_f32(
                false, a, false, bv, (short)0, acc, false, false);
        }

        if (tile + 1 < TILES_) async_wait0();   // next tile's DMA landed
        __syncthreads();
    }

    // ---- write this wave's 16x16 C tile to its private partial slot ----
    // C/D layout: lanes 0-15: VGPR j -> (M=j, N=lane); lanes 16-31: (M=j+8, N=lane-16)
    float* wp = wfp + (size_t)(b * NSPLIT_ + slab) * 16 * 128;
    const int mbase = (lane >> 4) * 8;
    const int c     = cw + (lane & 15);
    #pragma unroll
    for (int j = 0; j < 8; ++j) {
        wp[(size_t)(j + mbase) * 128 + c] = acc[j];
    }
}

// ---------------------------------------------------------------------------
// Pass 2: reduce the 32 slab partials, then contract with weights[k,c,o].
// One block per batch, 128 threads (= output channels). Deterministic order.
// ---------------------------------------------------------------------------
__global__ __launch_bounds__(128) void kpconv_finalize_kernel(
    const float* __restrict__ wfp,      // [B, NSPLIT, 16, 128]
    const float* __restrict__ wts,      // [K, CIN, COUT]
    float* __restrict__ out)            // [B, COUT]
{
    __shared__ float swf[16 * 128];
    const int b = blockIdx.x;
    const int t = threadIdx.x;          // output channel o

    const float* base = wfp + (size_t)b * NSPLIT_ * 16 * 128;
    #pragma unroll 1
    for (int i = 0; i < 16; ++i) {
        const int e = t + 128 * i;      // flat index into 16x128
        float s = 0.f;
        for (int sl = 0; sl < NSPLIT_; ++sl)
            s += base[(size_t)sl * 16 * 128 + e];
        swf[e] = s;
    }
    __syncthreads();

    float accv = 0.f;
    for (int k = 0; k < KP_; ++k) {
        const float* frow = swf + k * 128;
        const float* wrow = wts + (size_t)k * CIN_ * COUT_;
        #pragma unroll 4
        for (int c = 0; c < CIN_; ++c)
            accv = fmaf(frow[c], wrow[(size_t)c * COUT_ + t], accv);
    }
    out[b * COUT_ + t] = accv;
}

// ---------------------------------------------------------------------------
extern "C" void kernel_launch(void* const* d_in, const int* in_sizes, int n_in,
                              void* d_out, int out_size, void* d_ws, size_t ws_size,
                              hipStream_t stream)
{
    (void)in_sizes; (void)n_in; (void)out_size; (void)ws_size;
    const float* p   = (const float*)d_in[0];   // [B, N, 3]
    const float* x   = (const float*)d_in[1];   // [B, CIN, N]
    const float* wts = (const float*)d_in[2];   // [K, CIN, COUT]
    const float* kp  = (const float*)d_in[3];   // [K, 3]
    float* out = (float*)d_out;
    float* wfp = (float*)d_ws;                  // needs B*NSPLIT*16*128*4 = 2 MB

    kpconv_partial_kernel<<<B_ * NSPLIT_, 256, 0, stream>>>(p, x, kp, wfp);
    kpconv_finalize_kernel<<<B_, 128, 0, stream>>>(wfp, wts, out);
}